// RGB_HVI_30932354466502
// MI455X (gfx1250) — compile-verified
//
#include <hip/hip_runtime.h>

// RGB -> HVI -> RGB elementwise roundtrip for (16,3,1024,1024) fp32.
// Memory-bound: 402 MB traffic / 23.3 TB/s ~= 17 us floor. All transcendentals
// map to single CDNA5 hardware ops (v_sin/v_cos take revolutions: sin(2*pi*x)).
// Hot kernel is pure vector path; remainder (never hit for this shape) goes to
// a separate tail kernel so the hot loop stays small for the I$/prefetcher.

typedef float float4v __attribute__((ext_vector_type(4)));

#define EPSF 1e-8f

__device__ __forceinline__ float rcpf(float x) { return __builtin_amdgcn_rcpf(x); }

// atan2(y,x) in REVOLUTIONS (atan2/(2*pi)), range [-0.5, 0.5].
// Degree-11 minimax atan poly, max err ~1e-6 rad.
__device__ __forceinline__ float atan2_rev(float y, float x) {
    float ax = __builtin_fabsf(x), ay = __builtin_fabsf(y);
    float mx = fmaxf(ax, ay);
    float mn = fminf(ax, ay);
    float a  = mn * rcpf(fmaxf(mx, 1e-30f));   // guard (0,0) -> a = 0
    float a2 = a * a;
    float p = __builtin_fmaf(a2, -1.1721201e-2f, 5.2653320e-2f);
    p = __builtin_fmaf(a2, p, -1.1643287e-1f);
    p = __builtin_fmaf(a2, p,  1.9354346e-1f);
    p = __builtin_fmaf(a2, p, -3.3262347e-1f);
    p = __builtin_fmaf(a2, p,  9.9997726e-1f);
    float r = a * p;                            // atan(mn/mx) in [0, pi/4]
    r = (ay > ax) ? (1.57079632679f - r) : r;   // octant -> quadrant 1
    r = (x < 0.0f) ? (3.14159265359f - r) : r;  // quadrant 2
    r *= 0.15915494309f;                        // radians -> revolutions
    r = (y < 0.0f) ? -r : r;                    // lower half plane
    return r;
}

__device__ __forceinline__ void hvi_pixel(float r, float g, float b, float kk,
                                          float& ro, float& go, float& bo)
{
    // ---------------- forward: _hvit ----------------
    float value = fmaxf(r, fmaxf(g, b));
    float vmin  = fminf(r, fminf(g, b));
    float rdiff = rcpf(value - vmin + EPSF);

    // jnp.where chain, later writes override earlier ones:
    float hx = (g - b) * rdiff;                 // (g-b)/diff % 6.0  (x in (-6,6))
    hx = (hx < 0.0f) ? hx + 6.0f : hx;
    float hue = (b == value) ? __builtin_fmaf(r - g, rdiff, 4.0f) : 0.0f;
    hue = (g == value) ? __builtin_fmaf(b - r, rdiff, 2.0f) : hue;
    hue = (r == value) ? hx : hue;
    hue = (vmin == value) ? 0.0f : hue;
    hue *= (1.0f / 6.0f);                       // hue in [0,1) revolutions

    float sat = (value == 0.0f) ? 0.0f : (value - vmin) * rcpf(value + EPSF);

    // cs = (sin(value*pi/2) + eps)^k ; v_sin takes revolutions -> value/4
    float sv = __builtin_amdgcn_sinf(value * 0.25f) + EPSF;
    float cs = __builtin_amdgcn_exp2f(kk * __builtin_amdgcn_logf(sv));

    float X = cs * sat * __builtin_amdgcn_cosf(hue);   // cos(2*pi*hue)
    float Y = cs * sat * __builtin_amdgcn_sinf(hue);   // sin(2*pi*hue)

    // ---------------- inverse: _phvit ----------------
    float v  = fminf(fmaxf(value, 0.0f), 1.0f);
    float cb = cs;
    if (v != value) {   // wave-coherent; never taken for in-range inputs
        float sv2 = __builtin_amdgcn_sinf(v * 0.25f) + EPSF;
        cb = __builtin_amdgcn_exp2f(kk * __builtin_amdgcn_logf(sv2));
    }
    float H = fminf(fmaxf(X, -1.0f), 1.0f);
    float V = fminf(fmaxf(Y, -1.0f), 1.0f);
    float rc = rcpf(cb + EPSF);
    H = fminf(fmaxf(H * rc, -1.0f), 1.0f);
    V = fminf(fmaxf(V * rc, -1.0f), 1.0f);

    float h = atan2_rev(V, H);
    h = h - __builtin_floorf(h);                // mod 1.0
    float s = fminf(__builtin_amdgcn_sqrtf(__builtin_fmaf(H, H, V * V)), 1.0f);

    float h6 = h * 6.0f;
    float fl = __builtin_floorf(h6);
    float f  = h6 - fl;
    int   hi = (int)fl;

    float p = v * (1.0f - s);
    float q = v * (1.0f - f * s);
    float t = v * (1.0f - (1.0f - f) * s);

    ro = (hi==0)?v:(hi==1)?q:(hi==2)?p:(hi==3)?p:(hi==4)?t:(hi==5)?v:0.0f;
    go = (hi==0)?t:(hi==1)?v:(hi==2)?v:(hi==3)?q:(hi==4)?p:(hi==5)?p:0.0f;
    bo = (hi==0)?p:(hi==1)?p:(hi==2)?t:(hi==3)?v:(hi==4)?v:(hi==5)?q:0.0f;
}

// Hot path: exactly 4 pixels per thread, all in-bounds (host guarantees).
__global__ __launch_bounds__(256) void hvi_roundtrip_vec4(
    const float* __restrict__ img, const float* __restrict__ kptr,
    float* __restrict__ out, int nvec /* npix/4 */)
{
    constexpr int HW = 1 << 20;                 // 1024*1024 channel stride
    const float kk = kptr[0];                   // uniform scalar load

    int tid = blockIdx.x * blockDim.x + threadIdx.x;
    if (tid >= nvec) return;
    int q = tid * 4;

    int n = q >> 20;                            // image index
    int o = q & (HW - 1);                       // pixel within image (4-aligned)
    int base = n * (3 * HW) + o;

    // streaming, touched once -> non-temporal b128 loads/stores
    float4v r4 = __builtin_nontemporal_load((const float4v*)(img + base));
    float4v g4 = __builtin_nontemporal_load((const float4v*)(img + base + HW));
    float4v b4 = __builtin_nontemporal_load((const float4v*)(img + base + 2 * HW));
    float4v ro, go, bo;
#pragma unroll
    for (int i = 0; i < 4; ++i) {
        float rr, gg, bb;
        hvi_pixel(r4[i], g4[i], b4[i], kk, rr, gg, bb);
        ro[i] = rr; go[i] = gg; bo[i] = bb;
    }
    __builtin_nontemporal_store(ro, (float4v*)(out + base));
    __builtin_nontemporal_store(go, (float4v*)(out + base + HW));
    __builtin_nontemporal_store(bo, (float4v*)(out + base + 2 * HW));
}

// Remainder path (only launched if npix % 4 != 0; never for this shape).
__global__ __launch_bounds__(64) void hvi_roundtrip_tail(
    const float* __restrict__ img, const float* __restrict__ kptr,
    float* __restrict__ out, int start, int npix)
{
    constexpr int HW = 1 << 20;
    const float kk = kptr[0];
    int j = start + blockIdx.x * blockDim.x + threadIdx.x;
    if (j >= npix) return;
    int n = j >> 20, o = j & (HW - 1);
    int base = n * (3 * HW) + o;
    float rr, gg, bb;
    hvi_pixel(img[base], img[base + HW], img[base + 2 * HW], kk, rr, gg, bb);
    out[base] = rr; out[base + HW] = gg; out[base + 2 * HW] = bb;
}

extern "C" void kernel_launch(void* const* d_in, const int* in_sizes, int n_in,
                              void* d_out, int out_size, void* d_ws, size_t ws_size,
                              hipStream_t stream) {
    (void)n_in; (void)out_size; (void)d_ws; (void)ws_size;
    const float* img  = (const float*)d_in[0];
    const float* kptr = (const float*)d_in[1];
    float* out = (float*)d_out;

    const int npix = in_sizes[0] / 3;           // 16 * 1024 * 1024
    const int nvec = npix / 4;
    const int block = 256;
    const int grid = (nvec + block - 1) / block;

    if (nvec > 0) {
        hipLaunchKernelGGL(hvi_roundtrip_vec4, dim3(grid), dim3(block), 0, stream,
                           img, kptr, out, nvec);
    }
    const int rem = npix - nvec * 4;
    if (rem > 0) {  // deterministic function of input sizes; graph-capture safe
        hipLaunchKernelGGL(hvi_roundtrip_tail, dim3(1), dim3(64), 0, stream,
                           img, kptr, out, nvec * 4, npix);
    }
}